// MultiResHashGrid_4054449128091
// MI455X (gfx1250) — compile-verified
//
#include <hip/hip_runtime.h>
#include <stdint.h>

typedef int   v4i __attribute__((vector_size(16)));   // matches builtin param type
typedef float v4f __attribute__((ext_vector_type(4)));
#define AS1 __attribute__((address_space(1)))
#define AS3 __attribute__((address_space(3)))

#define NLEVELS 16
#define TPAD    (1u << 19)   // padded table entries per level (2^LOG2_T)
#define BLK     256
#define ROWPAD  36           // 32 outputs + 4 pad floats; 144 B rows stay 16B-aligned
#define L0CAP   4096         // LDS-staged level-0 table capacity (entries, 32 KB)

#define PRIME1  2654435761u
#define PRIME2  805459861u

// pointer casts for the async builtins (addrspace moves go through integers;
// low 32 bits of a generic LDS address are the LDS byte offset)
__device__ __forceinline__ AS1 v4i* g4(const void* p) {
    return (AS1 v4i*)(uintptr_t)p;
}
__device__ __forceinline__ AS3 v4i* l4(const void* p) {
    return (AS3 v4i*)(unsigned int)(uintptr_t)p;
}

// one hash-grid level: 8 corner gathers + trilinear blend.
// tbl may point to LDS (after inlining clang propagates addrspace -> ds_load_b64)
// or to global memory (global_load_b64).
__device__ __forceinline__ void interp_level(float sx, float sy, float sz,
                                             unsigned int m, unsigned long long M,
                                             const float2* __restrict__ tbl,
                                             float& o0r, float& o1r)
{
    const float fxi = floorf(sx), fyi = floorf(sy), fzi = floorf(sz);
    const int ix = (int)fxi, iy = (int)fyi, iz = (int)fzi;
    const float fx = sx - fxi, fy = sy - fyi, fz = sz - fzi;

    // per-dim hash terms (uint32 wraparound == jnp.uint32 semantics; prime[0]==1)
    const unsigned int hx0 = (unsigned int)ix;
    const unsigned int hx1 = (unsigned int)(ix + 1);
    const unsigned int hy0 = (unsigned int)iy * PRIME1;
    const unsigned int hy1 = (unsigned int)(iy + 1) * PRIME1;
    const unsigned int hz0 = (unsigned int)iz * PRIME2;
    const unsigned int hz1 = (unsigned int)(iz + 1) * PRIME2;

    const float wx0 = 1.0f - fx, wy0 = 1.0f - fy, wz0 = 1.0f - fz;

    float o0 = 0.0f, o1 = 0.0f;
    if (M == 0ull) {                 // power-of-two table: modulo is an AND
        const unsigned int mask = m - 1u;
#pragma unroll
        for (int c = 0; c < 8; ++c) {
            const unsigned int h = ((c & 1) ? hx1 : hx0)
                                 ^ ((c & 2) ? hy1 : hy0)
                                 ^ ((c & 4) ? hz1 : hz0);
            const float2 e = tbl[h & mask];
            const float w = (((c & 1) ? fx : wx0)
                           * ((c & 2) ? fy : wy0))
                           * ((c & 4) ? fz : wz0);
            o0 = fmaf(w, e.x, o0);
            o1 = fmaf(w, e.y, o1);
        }
    } else {                         // exact h % m via Lemire fastmod (2 wide muls)
#pragma unroll
        for (int c = 0; c < 8; ++c) {
            const unsigned int h = ((c & 1) ? hx1 : hx0)
                                 ^ ((c & 2) ? hy1 : hy0)
                                 ^ ((c & 4) ? hz1 : hz0);
            const unsigned long long lowbits = M * (unsigned long long)h;
            const unsigned int j = (unsigned int)__umul64hi(lowbits, (unsigned long long)m);
            const float2 e = tbl[j];
            const float w = (((c & 1) ? fx : wx0)
                           * ((c & 2) ? fy : wy0))
                           * ((c & 4) ? fz : wz0);
            o0 = fmaf(w, e.x, o0);
            o1 = fmaf(w, e.y, o1);
        }
    }
    o0r = o0; o1r = o1;
}

__global__ __launch_bounds__(BLK)
void hashgrid_fwd_kernel(const float* __restrict__ x,
                         const float* __restrict__ tables,
                         const int*   __restrict__ resolutions,
                         const int*   __restrict__ hsizes,
                         float*       __restrict__ out,
                         int npoints)
{
    __shared__ unsigned long long s_magic[NLEVELS];  // 0 => power-of-two, use mask
    __shared__ float              s_res[NLEVELS];
    __shared__ unsigned int       s_hs[NLEVELS];
    __shared__ float2             s_tab0[L0CAP];       // 32 KB staged level-0 table
    __shared__ float              s_out[BLK * ROWPAD]; // 36 KB output transpose tile

    const int tid = threadIdx.x;

    // ---- kick off async DMA of the level-0 table into LDS (ASYNCcnt path) ----
    const unsigned int hs0 = (unsigned int)hsizes[0];          // uniform s_load
    const bool stage0 = (hs0 <= (unsigned int)L0CAP);
    if (stage0) {
        const int nchunks = (int)((hs0 * 8u + 15u) / 16u);     // 16B chunks
        const char* gsrc = (const char*)tables;                // level 0 at offset 0
        const char* ldst = (const char*)s_tab0;
        for (int i = tid; i < nchunks; i += BLK) {
            __builtin_amdgcn_global_load_async_to_lds_b128(
                g4(gsrc + (size_t)i * 16),
                l4(ldst + i * 16),
                0 /*offset*/, 0 /*cpol RT*/);
        }
    }

    // ---- per-block level params + Lemire magics (16 u64 divides / block) ----
    if (tid < NLEVELS) {
        unsigned int m = (unsigned int)hsizes[tid];
        s_hs[tid]  = m;
        s_res[tid] = (float)resolutions[tid];
        s_magic[tid] = ((m & (m - 1u)) == 0u)
                         ? 0ull
                         : (0xFFFFFFFFFFFFFFFFull / (unsigned long long)m + 1ull);
    }
    __syncthreads();

    const int p = blockIdx.x * BLK + tid;
    float x0 = 0.f, x1 = 0.f, x2 = 0.f;
    float* __restrict__ row = s_out + tid * ROWPAD;

    if (p < npoints) {
        // x is read exactly once: non-temporal loads
        x0 = __builtin_nontemporal_load(x + 3 * (size_t)p + 0);
        x1 = __builtin_nontemporal_load(x + 3 * (size_t)p + 1);
        x2 = __builtin_nontemporal_load(x + 3 * (size_t)p + 2);

        // ---- levels 1..15 from global (L2-resident tables) while DMA is in flight.
        //      unroll 3 (15 = 3x5): 24 independent gathers in flight per iteration.
#pragma unroll 3
        for (int l = 1; l < NLEVELS; ++l) {
            const float r = s_res[l];
            float o0, o1;
            interp_level(x0 * r, x1 * r, x2 * r, s_hs[l], s_magic[l],
                         (const float2*)tables + (size_t)l * TPAD, o0, o1);
            row[2 * l + 0] = o0;
            row[2 * l + 1] = o1;
        }
    }

    // ---- drain the table DMA, then serve level 0 from LDS ----
    __builtin_amdgcn_s_wait_asynccnt(0);
    __syncthreads();

    if (p < npoints) {
        const float r = s_res[0];
        float o0, o1;
        if (stage0) {
            interp_level(x0 * r, x1 * r, x2 * r, s_hs[0], s_magic[0],
                         (const float2*)s_tab0, o0, o1);        // ds_load_b64 gathers
        } else {
            interp_level(x0 * r, x1 * r, x2 * r, s_hs[0], s_magic[0],
                         (const float2*)tables, o0, o1);
        }
        row[0] = o0;
        row[1] = o1;
    }
    __syncthreads();

    // ---- coalesced output: async store LDS -> global, non-temporal ----
    // Block tile: 256 points * 32 floats = 2048 b128 chunks, fully coalesced.
    const size_t blockbase = (size_t)blockIdx.x * BLK * 32;
    char* gdst_base = (char*)out;
#pragma unroll
    for (int k = 0; k < 8; ++k) {
        const int q  = k * BLK + tid;         // b128 index within tile, 0..2047
        const int pl = q >> 3;                // local point
        if (blockIdx.x * BLK + pl < npoints) {
            const int col = (q & 7) * 4;      // float column within the 32-float row
            const float* src = s_out + pl * ROWPAD + col;
            __builtin_amdgcn_global_store_async_from_lds_b128(
                g4(gdst_base + (blockbase + (size_t)q * 4) * sizeof(float)),
                l4(src),
                0 /*offset*/, 1 /*cpol TH_NT*/);
        }
    }
    // S_ENDPGM performs an implicit wait-idle, draining ASYNCcnt before LDS dealloc.
}

extern "C" void kernel_launch(void* const* d_in, const int* in_sizes, int n_in,
                              void* d_out, int out_size, void* d_ws, size_t ws_size,
                              hipStream_t stream)
{
    const float* x      = (const float*)d_in[0];
    const float* tables = (const float*)d_in[1];
    const int*   res    = (const int*)d_in[2];
    const int*   hs     = (const int*)d_in[3];
    float*       out    = (float*)d_out;

    const int npoints = in_sizes[0] / 3;
    const int blocks  = (npoints + BLK - 1) / BLK;

    hipLaunchKernelGGL(hashgrid_fwd_kernel, dim3(blocks), dim3(BLK), 0, stream,
                       x, tables, res, hs, out, npoints);
}